// lct_34205119545600
// MI455X (gfx1250) — compile-verified
//
#include <hip/hip_runtime.h>

typedef __attribute__((ext_vector_type(2))) float v2f;
typedef __attribute__((ext_vector_type(8))) float v8f;
typedef __attribute__((ext_vector_type(2))) int   v2i;

namespace {
constexpr int BD = 2;                 // b * d
constexpr int T  = 256, H  = 128, W  = 128;
constexpr int T2 = 512, H2 = 256, W2 = 256;
constexpr long long VOL2      = (long long)T2 * H2 * W2;      // 33,554,432
constexpr long long PAD_ELEMS = (long long)BD * VOL2;         // 67,108,864 complex
}

// ---------------------------------------------------------------------------
// CDNA5 async global<->LDS b64 copies (ASYNCcnt-tracked, bypass VGPRs).
// Builtin signature (from hipcc diagnostic): global ptr is int2* in target
// AS1 ("__device__"); LDS ptr in AS3. Pointers built via int round-trips:
// LDS byte offset = low 32 bits of the flat address (LDS aperture rule).
// ---------------------------------------------------------------------------
typedef __attribute__((address_space(1))) v2i as1_v2i;
typedef __attribute__((address_space(3))) v2i as3_v2i;

__device__ __forceinline__ void async_g2l_b64(const void* gsrc, void* ldst)
{
#if __has_builtin(__builtin_amdgcn_global_load_async_to_lds_b64)
    __builtin_amdgcn_global_load_async_to_lds_b64(
        (as1_v2i*)(unsigned long long)gsrc,
        (as3_v2i*)(unsigned long long)(unsigned)(unsigned long long)ldst,
        0, 0);
#else
    asm volatile("global_load_async_to_lds_b64 %0, %1, off"
                 :: "v"((unsigned)(unsigned long long)ldst), "v"(gsrc)
                 : "memory");
#endif
}

__device__ __forceinline__ void async_l2g_b64(void* gdst, const void* lsrc)
{
#if __has_builtin(__builtin_amdgcn_global_store_async_from_lds_b64)
    __builtin_amdgcn_global_store_async_from_lds_b64(
        (as1_v2i*)(unsigned long long)gdst,
        (as3_v2i*)(unsigned long long)(unsigned)(unsigned long long)lsrc,
        0, 0);
#else
    asm volatile("global_store_async_from_lds_b64 %0, %1, off"
                 :: "v"(gdst), "v"((unsigned)(unsigned long long)lsrc)
                 : "memory");
#endif
}

__device__ __forceinline__ void wait_asynccnt0()
{
#if __has_builtin(__builtin_amdgcn_s_wait_asynccnt)
    __builtin_amdgcn_s_wait_asynccnt(0);
#else
    asm volatile("s_wait_asynccnt 0x0" ::: "memory");
#endif
}

// ---------------------------------------------------------------------------
// Front matmul: pad[bd, m, h, w] = sum_t mtx[m,t] * (feature[bd,t,h,w] * gz[t]^4)
// One wave computes a 16(m) x 16(col) tile via V_WMMA_F32_16X16X4_F32, K=256.
// A (16x4 f32): lanes 0-15 -> M rows, lane[4] selects K pair {0,1}/{2,3}.
// B (4x16 f32): lanes 0-15 -> N cols (mirrored layout).
// C/D (16x16 f32, 8 VGPRs): vgpr r: lanes0-15 M=r, lanes16-31 M=r+8.
// ---------------------------------------------------------------------------
__global__ __launch_bounds__(32)
void lct_front_mm(const float* __restrict__ feature, const float* __restrict__ gridz,
                  const float* __restrict__ mtx, float2* __restrict__ pad)
{
    const int lane  = threadIdx.x & 31;
    const int nIdx  = lane & 15;
    const int kHalf = lane >> 4;              // 0 or 1
    const int col0  = blockIdx.x * 16;        // column within H*W (tile fits one h-row)
    const int m0    = blockIdx.y * 16;
    const int bd    = blockIdx.z;

    const float* fbase = feature + (size_t)bd * T * H * W + col0;
    v8f acc = {};
    for (int k0 = 0; k0 < T; k0 += 4) {
        const int t0 = k0 + 2 * kHalf;
        v2f a, b;
        a.x = mtx[(m0 + nIdx) * T + t0];
        a.y = mtx[(m0 + nIdx) * T + t0 + 1];
        float g0 = gridz[t0];     g0 *= g0; g0 *= g0;   // gz^4
        float g1 = gridz[t0 + 1]; g1 *= g1; g1 *= g1;
        b.x = fbase[(size_t)t0       * (H * W) + nIdx] * g0;
        b.y = fbase[(size_t)(t0 + 1) * (H * W) + nIdx] * g1;
        acc = __builtin_amdgcn_wmma_f32_16x16x4_f32(false, a, false, b,
                                                    (short)0, acc, false, false);
    }
    const int h = col0 / W;
    const int w = (col0 % W) + nIdx;
#pragma unroll
    for (int r = 0; r < 8; ++r) {
        const int m = m0 + r + 8 * kHalf;
        pad[(((size_t)bd * T2 + m) * H2 + h) * W2 + w] = make_float2(acc[r], 0.0f);
    }
}

// ---------------------------------------------------------------------------
// Back matmul (fused crop): out[bd, m, h, w] = sum_t mtxi[m,t] * Re(pad[bd,t,h,w])
// ---------------------------------------------------------------------------
__global__ __launch_bounds__(32)
void lct_back_mm(const float2* __restrict__ pad, const float* __restrict__ mtxi,
                 float* __restrict__ out)
{
    const int lane  = threadIdx.x & 31;
    const int nIdx  = lane & 15;
    const int kHalf = lane >> 4;
    const int col0  = blockIdx.x * 16;
    const int m0    = blockIdx.y * 16;
    const int bd    = blockIdx.z;

    const int h = col0 / W;
    const int w = (col0 % W) + nIdx;
    v8f acc = {};
    for (int k0 = 0; k0 < T; k0 += 4) {
        const int t0 = k0 + 2 * kHalf;
        v2f a, b;
        a.x = mtxi[(m0 + nIdx) * T + t0];
        a.y = mtxi[(m0 + nIdx) * T + t0 + 1];
        b.x = pad[(((size_t)bd * T2 + t0)     * H2 + h) * W2 + w].x;
        b.y = pad[(((size_t)bd * T2 + t0 + 1) * H2 + h) * W2 + w].x;
        acc = __builtin_amdgcn_wmma_f32_16x16x4_f32(false, a, false, b,
                                                    (short)0, acc, false, false);
    }
#pragma unroll
    for (int r = 0; r < 8; ++r) {
        const int m = m0 + r + 8 * kHalf;
        out[(((size_t)bd * T + m) * H + h) * W + w] = acc[r];
    }
}

// ---------------------------------------------------------------------------
// Batched 1D Stockham radix-2 FFT in LDS (ping-pong, autosort -> no bit rev).
// Each block transforms TILE lines of length N. Tile staged into LDS with
// CDNA5 async global->LDS b64 copies; results drained with async LDS->global.
//   element addr = base + line*line_stride + pos*pos_stride
//   base         = (g/div)*strideA + (g%div)*strideB
// CONTIG=true : lines contiguous in memory (W axis); LDS layout lds[line][pos]
// CONTIG=false: strided axis, TILE adjacent lines coalesce; lds[pos][line]
// DIR = -1 forward, +1 inverse (1/N folded into the filter multiply).
// ---------------------------------------------------------------------------
template<int N, int TILE, bool CONTIG, int DIR>
__global__ __launch_bounds__(256)
void lct_fft(float2* __restrict__ data, int div, long long strideA, long long strideB,
             long long line_stride, long long pos_stride)
{
    __shared__ float2 ldsA[N * TILE];
    __shared__ float2 ldsB[N * TILE];
    constexpr int TOT  = N * TILE;
    constexpr int HALF = (N / 2) * TILE;
    constexpr int P    = (N == 512) ? 9 : 8;
    constexpr float CW = 6.28318530717958647692f / (float)N;   // 2*pi/N

    const int tid = threadIdx.x;
    const int g   = blockIdx.x;
    const size_t base = (size_t)(g / div) * (size_t)strideA +
                        (size_t)(g % div) * (size_t)strideB;

    // Stage tile into LDS via async copies (coalesced, VGPR-free)
#pragma unroll 1
    for (int e = tid; e < TOT; e += 256) {
        int line, pos;
        if (CONTIG) { line = e / N;    pos = e - line * N; }
        else        { line = e % TILE; pos = e / TILE;     }
        const int li = CONTIG ? (line * N + pos) : (pos * TILE + line);
        async_g2l_b64(&data[base + (size_t)line * line_stride +
                                   (size_t)pos * pos_stride], &ldsA[li]);
    }
    wait_asynccnt0();
    __syncthreads();

    float2* src = ldsA;
    float2* dst = ldsB;
#pragma unroll 1
    for (int s = 0; s < P; ++s) {
        const int m = 1 << s;
#pragma unroll 1
        for (int f = tid; f < HALF; f += 256) {
            int line, bf;
            if (CONTIG) { line = f / (N / 2); bf = f - line * (N / 2); }
            else        { line = f % TILE;    bf = f / TILE;           }
            const int k       = bf & (m - 1);
            const int lineOff = CONTIG ? (line * N) : line;
            const int mulc    = CONTIG ? 1 : TILE;
            const float2 a = src[lineOff + bf * mulc];
            const float2 b = src[lineOff + (bf + N / 2) * mulc];
            // twiddle: exp(-i*2pi*j*m/N), j*m == bf & ~(m-1)  (angle in [0, pi))
            const float ang = CW * (float)(bf - k);
            float sn, cs;
            __sincosf(ang, &sn, &cs);
            const float wr = cs;
            const float wi = (DIR < 0) ? -sn : sn;
            const float2 sum = make_float2(a.x + b.x, a.y + b.y);
            const float2 dif = make_float2(a.x - b.x, a.y - b.y);
            const float2 tw  = make_float2(dif.x * wr - dif.y * wi,
                                           dif.x * wi + dif.y * wr);
            dst[lineOff + (2 * bf - k) * mulc]     = sum;
            dst[lineOff + (2 * bf - k + m) * mulc] = tw;
        }
        __syncthreads();
        float2* t2 = src; src = dst; dst = t2;
    }

    // Drain tile (result lives in src after final swap) via async LDS->global
#pragma unroll 1
    for (int e = tid; e < TOT; e += 256) {
        int line, pos;
        if (CONTIG) { line = e / N;    pos = e - line * N; }
        else        { line = e % TILE; pos = e / TILE;     }
        const int li = CONTIG ? (line * N + pos) : (pos * TILE + line);
        async_l2g_b64(&data[base + (size_t)line * line_stride +
                                   (size_t)pos * pos_stride], &src[li]);
    }
    wait_asynccnt0();
}

// ---------------------------------------------------------------------------
// Pointwise Wiener filter: pad *= invpsf * (1/VOL2); the inverse-FFT 1/N is
// folded in here. invpsf broadcast over the BD channels. 2 complex / iter.
// ---------------------------------------------------------------------------
__global__ __launch_bounds__(256)
void lct_cmul(float4* __restrict__ pad4, const float2* __restrict__ pr2,
              const float2* __restrict__ pi2)
{
    const float scale = 1.0f / (float)VOL2;     // 2^-25, exact
    const long long pairs = PAD_ELEMS / 2;
    for (long long p = (long long)blockIdx.x * blockDim.x + threadIdx.x;
         p < pairs; p += (long long)gridDim.x * blockDim.x) {
        const long long j = p & (VOL2 / 2 - 1);
        const float4 v = pad4[p];
        const float2 wr = pr2[j];
        const float2 wi = pi2[j];
        pad4[p] = make_float4((v.x * wr.x - v.y * wi.x) * scale,
                              (v.x * wi.x + v.y * wr.x) * scale,
                              (v.z * wr.y - v.w * wi.y) * scale,
                              (v.z * wi.y + v.w * wr.y) * scale);
    }
}

// ---------------------------------------------------------------------------
extern "C" void kernel_launch(void* const* d_in, const int* in_sizes, int n_in,
                              void* d_out, int out_size, void* d_ws, size_t ws_size,
                              hipStream_t stream)
{
    (void)in_sizes; (void)n_in; (void)out_size; (void)ws_size;
    const float* feature = (const float*)d_in[0];   // [1,2,256,128,128]
    const float* gridz   = (const float*)d_in[1];   // [1,256,1,1]
    const float* mtx     = (const float*)d_in[2];   // [256,256]
    const float* mtxi    = (const float*)d_in[3];   // [256,256]
    const float* pr      = (const float*)d_in[4];   // [1,512,256,256]
    const float* pi      = (const float*)d_in[5];
    float*  out = (float*)d_out;
    float2* pad = (float2*)d_ws;                    // 536,870,912 bytes

    (void)hipMemsetAsync(pad, 0, (size_t)PAD_ELEMS * sizeof(float2), stream);

    const dim3 mmGrid(H * W / 16, T / 16, BD);      // (1024, 16, 2)
    lct_front_mm<<<mmGrid, 32, 0, stream>>>(feature, gridz, mtx, pad);

    // Forward FFT: W (contiguous), H, T
    lct_fft<256, 8, true,  -1><<<BD * T2 * H2 / 8, 256, 0, stream>>>(
        pad, 1, 8LL * W2, 0, W2, 1);
    lct_fft<256, 8, false, -1><<<BD * T2 * (W2 / 8), 256, 0, stream>>>(
        pad, W2 / 8, (long long)H2 * W2, 8, 1, W2);
    lct_fft<512, 4, false, -1><<<BD * (H2 * W2 / 4), 256, 0, stream>>>(
        pad, H2 * W2 / 4, (long long)T2 * H2 * W2, 4, 1, (long long)H2 * W2);

    lct_cmul<<<32768, 256, 0, stream>>>((float4*)pad, (const float2*)pr,
                                        (const float2*)pi);

    // Inverse FFT: T, H, W
    lct_fft<512, 4, false, 1><<<BD * (H2 * W2 / 4), 256, 0, stream>>>(
        pad, H2 * W2 / 4, (long long)T2 * H2 * W2, 4, 1, (long long)H2 * W2);
    lct_fft<256, 8, false, 1><<<BD * T2 * (W2 / 8), 256, 0, stream>>>(
        pad, W2 / 8, (long long)H2 * W2, 8, 1, W2);
    lct_fft<256, 8, true,  1><<<BD * T2 * H2 / 8, 256, 0, stream>>>(
        pad, 1, 8LL * W2, 0, W2, 1);

    lct_back_mm<<<mmGrid, 32, 0, stream>>>(pad, mtxi, out);
}